// OmicGATEncoder_66907000537156
// MI455X (gfx1250) — compile-verified
//
#include <hip/hip_runtime.h>

#define N_NODES 20000
#define HIDDEN  128
#define HEADS   8
#define NEG_SLOPE 0.2f

typedef __attribute__((ext_vector_type(2))) float v2f;
typedef __attribute__((ext_vector_type(8))) float v8f;

// ---- order-preserving float<->uint encoding so atomicMax works on signed floats ----
__device__ __forceinline__ unsigned ord_enc(float f) {
  unsigned u = __float_as_uint(f);
  return u ^ ((unsigned)((int)u >> 31) | 0x80000000u);
}
__device__ __forceinline__ float ord_dec(unsigned e) {
  unsigned u = e ^ ((unsigned)((int)(~e) >> 31) | 0x80000000u);
  return __uint_as_float(u);
}

// C[M x 128] = A[M x K] @ B[K x 128] + bias, all row-major, full f32 via WMMA.
// One wave computes one 16x16 output tile; 8 waves/block cover all 128 columns.
__global__ void gemm_bias_f32wmma(const float* __restrict__ A,
                                  const float* __restrict__ B,
                                  const float* __restrict__ bias,
                                  float* __restrict__ C,
                                  int M, int K) {
  const int wave = threadIdx.x >> 5;
  const int lane = threadIdx.x & 31;
  const int m0 = blockIdx.x << 4;
  const int n0 = wave << 4;
  const int r15 = lane & 15;
  const int hi  = lane >> 4;               // lane half selects K pair {0,1} vs {2,3}
  v8f acc = {};
  const float* Arow = A + (size_t)(m0 + r15) * K;
  const float* Bcol = B + n0 + r15;
  #pragma unroll 4
  for (int kb = 0; kb < K; kb += 4) {
    const int ka = kb + (hi << 1);
    v2f a, b;
    a[0] = Arow[ka];
    a[1] = Arow[ka + 1];
    b[0] = Bcol[(size_t)ka * HIDDEN];
    b[1] = Bcol[(size_t)(ka + 1) * HIDDEN];
    acc = __builtin_amdgcn_wmma_f32_16x16x4_f32(false, a, false, b,
                                                (short)0, acc, false, false);
  }
  const int col = n0 + r15;
  const float bc = bias[col];
  #pragma unroll
  for (int r = 0; r < 8; ++r) {
    const int row = m0 + r + (hi << 3);    // C/D layout: VGPR r -> rows r and r+8
    C[(size_t)row * HIDDEN + col] = acc[r] + bc;
  }
}

// One wave per node: LayerNorm over 128 features then ReLU.
__global__ void ln_relu_node(const float* __restrict__ X,
                             const float* __restrict__ g,
                             const float* __restrict__ b,
                             float* __restrict__ H, int M) {
  const int gw = (int)((blockIdx.x * (size_t)blockDim.x + threadIdx.x) >> 5);
  const int lane = threadIdx.x & 31;
  if (gw >= M) return;
  const float4 v = *(const float4*)(X + (size_t)gw * HIDDEN + lane * 4);
  float s = v.x + v.y + v.z + v.w;
  float q = v.x * v.x + v.y * v.y + v.z * v.z + v.w * v.w;
  #pragma unroll
  for (int m = 1; m < 32; m <<= 1) {
    s += __shfl_xor(s, m, 32);
    q += __shfl_xor(q, m, 32);
  }
  const float mu  = s * (1.0f / 128.0f);
  const float var = q * (1.0f / 128.0f) - mu * mu;
  const float rs  = rsqrtf(var + 1e-5f);
  const float4 gg = *(const float4*)(g + lane * 4);
  const float4 bb = *(const float4*)(b + lane * 4);
  float4 o;
  o.x = fmaxf((v.x - mu) * rs * gg.x + bb.x, 0.0f);
  o.y = fmaxf((v.y - mu) * rs * gg.y + bb.y, 0.0f);
  o.z = fmaxf((v.z - mu) * rs * gg.z + bb.z, 0.0f);
  o.w = fmaxf((v.w - mu) * rs * gg.w + bb.w, 0.0f);
  *(float4*)(H + (size_t)gw * HIDDEN + lane * 4) = o;
}

__global__ void init_seg(float* __restrict__ accum,
                         unsigned* __restrict__ menc,
                         float* __restrict__ ssum) {
  const size_t idx = blockIdx.x * (size_t)blockDim.x + threadIdx.x;
  if (idx < (size_t)N_NODES * HIDDEN) accum[idx] = 0.0f;
  if (idx < (size_t)N_NODES * HEADS) {
    menc[idx] = 0x007FFFFFu;   // ord_enc(-inf)
    ssum[idx] = 0.0f;
  }
}

// One wave per edge: logits[e][h] = att[h] . leaky_relu(xl[src]+xr[dst]); atomic seg-max.
__global__ void edge_logits(const int* __restrict__ ei,
                            const float* __restrict__ xl,
                            const float* __restrict__ xr,
                            const float* __restrict__ att,
                            float* __restrict__ logits,
                            unsigned* __restrict__ menc,
                            int E) {
  const int gw = (int)((blockIdx.x * (size_t)blockDim.x + threadIdx.x) >> 5);
  const int lane = threadIdx.x & 31;
  const int Etot = E + N_NODES;
  if (gw >= Etot) return;
  int sn, dn;
  if (gw < E) { sn = ei[gw]; dn = ei[E + gw]; } else { sn = dn = gw - E; }
  const float4 a = *(const float4*)(xl + (size_t)sn * HIDDEN + lane * 4);
  const float4 c = *(const float4*)(xr + (size_t)dn * HIDDEN + lane * 4);
  const float4 w = *(const float4*)(att + lane * 4);
  float p = 0.0f, t;
  t = a.x + c.x; t = t > 0.0f ? t : NEG_SLOPE * t; p += t * w.x;
  t = a.y + c.y; t = t > 0.0f ? t : NEG_SLOPE * t; p += t * w.y;
  t = a.z + c.z; t = t > 0.0f ? t : NEG_SLOPE * t; p += t * w.z;
  t = a.w + c.w; t = t > 0.0f ? t : NEG_SLOPE * t; p += t * w.w;
  p += __shfl_xor(p, 1, 32);       // 4 lanes per head (16 channels / 4 per lane)
  p += __shfl_xor(p, 2, 32);
  if ((lane & 3) == 0) {
    const int h = lane >> 2;
    logits[(size_t)gw * HEADS + h] = p;
    atomicMax(&menc[(size_t)dn * HEADS + h], ord_enc(p));
  }
}

// One thread per (edge, head): ex = exp(logit - max[dst]); atomic seg-sum. In-place.
__global__ void edge_exp(const int* __restrict__ ei,
                         float* __restrict__ exbuf,
                         const unsigned* __restrict__ menc,
                         float* __restrict__ ssum, int E) {
  const size_t idx = blockIdx.x * (size_t)blockDim.x + threadIdx.x;
  const size_t tot = (size_t)(E + N_NODES) * HEADS;
  if (idx >= tot) return;
  const int e = (int)(idx >> 3);
  const int h = (int)(idx & 7);
  const int dn = (e < E) ? ei[E + e] : (e - E);
  const float m = ord_dec(menc[(size_t)dn * HEADS + h]);
  const float ex = expf(exbuf[idx] - m);
  exbuf[idx] = ex;
  atomicAdd(&ssum[(size_t)dn * HEADS + h], ex);
}

// One wave per edge: accum[dst] += xl[src] * alpha (per-head), float atomics.
__global__ void edge_scatter(const int* __restrict__ ei,
                             const float* __restrict__ xl,
                             const float* __restrict__ exbuf,
                             const float* __restrict__ ssum,
                             float* __restrict__ accum, int E) {
  const int gw = (int)((blockIdx.x * (size_t)blockDim.x + threadIdx.x) >> 5);
  const int lane = threadIdx.x & 31;
  const int Etot = E + N_NODES;
  if (gw >= Etot) return;
  int sn, dn;
  if (gw < E) { sn = ei[gw]; dn = ei[E + gw]; } else { sn = dn = gw - E; }
  const int h = lane >> 2;
  const float alpha = exbuf[(size_t)gw * HEADS + h] / ssum[(size_t)dn * HEADS + h];
  const float4 v = *(const float4*)(xl + (size_t)sn * HIDDEN + lane * 4);
  float* dp = accum + (size_t)dn * HIDDEN + lane * 4;
  atomicAdd(dp + 0, v.x * alpha);
  atomicAdd(dp + 1, v.y * alpha);
  atomicAdd(dp + 2, v.z * alpha);
  atomicAdd(dp + 3, v.w * alpha);
}

// One wave per node: out = accum + gat_bias; h = LN(elu(out) + h_in).
__global__ void node_finalize(const float* __restrict__ accum,
                              const float* __restrict__ gbias,
                              const float* __restrict__ g,
                              const float* __restrict__ b,
                              const float* __restrict__ Hin,
                              float* __restrict__ Hout, int M) {
  const int gw = (int)((blockIdx.x * (size_t)blockDim.x + threadIdx.x) >> 5);
  const int lane = threadIdx.x & 31;
  if (gw >= M) return;
  const float4 a  = *(const float4*)(accum + (size_t)gw * HIDDEN + lane * 4);
  const float4 gb = *(const float4*)(gbias + lane * 4);
  const float4 hv = *(const float4*)(Hin + (size_t)gw * HIDDEN + lane * 4);
  float4 v; float o;
  o = a.x + gb.x; o = o > 0.0f ? o : (expf(o) - 1.0f); v.x = o + hv.x;
  o = a.y + gb.y; o = o > 0.0f ? o : (expf(o) - 1.0f); v.y = o + hv.y;
  o = a.z + gb.z; o = o > 0.0f ? o : (expf(o) - 1.0f); v.z = o + hv.z;
  o = a.w + gb.w; o = o > 0.0f ? o : (expf(o) - 1.0f); v.w = o + hv.w;
  float s = v.x + v.y + v.z + v.w;
  float q = v.x * v.x + v.y * v.y + v.z * v.z + v.w * v.w;
  #pragma unroll
  for (int m = 1; m < 32; m <<= 1) {
    s += __shfl_xor(s, m, 32);
    q += __shfl_xor(q, m, 32);
  }
  const float mu  = s * (1.0f / 128.0f);
  const float var = q * (1.0f / 128.0f) - mu * mu;
  const float rs  = rsqrtf(var + 1e-5f);
  const float4 gg = *(const float4*)(g + lane * 4);
  const float4 bb = *(const float4*)(b + lane * 4);
  float4 out;
  out.x = (v.x - mu) * rs * gg.x + bb.x;
  out.y = (v.y - mu) * rs * gg.y + bb.y;
  out.z = (v.z - mu) * rs * gg.z + bb.z;
  out.w = (v.w - mu) * rs * gg.w + bb.w;
  *(float4*)(Hout + (size_t)gw * HIDDEN + lane * 4) = out;
}

extern "C" void kernel_launch(void* const* d_in, const int* in_sizes, int n_in,
                              void* d_out, int out_size, void* d_ws, size_t ws_size,
                              hipStream_t stream) {
  (void)n_in; (void)out_size; (void)ws_size;
  const float* x     = (const float*)d_in[0];
  const int*   ei    = (const int*)d_in[1];
  const float* W_in  = (const float*)d_in[2];
  const float* b_in  = (const float*)d_in[3];
  const float* lng   = (const float*)d_in[4];
  const float* lnb   = (const float*)d_in[5];
  const float* Wl    = (const float*)d_in[6];
  const float* bl    = (const float*)d_in[7];
  const float* Wr    = (const float*)d_in[8];
  const float* br    = (const float*)d_in[9];
  const float* att   = (const float*)d_in[10];
  const float* gbias = (const float*)d_in[11];
  const float* ng    = (const float*)d_in[12];
  const float* nb    = (const float*)d_in[13];

  const int M = N_NODES;
  const int E = in_sizes[1] / 2;          // 320000
  const int Etot = E + M;                 // + self loops
  const size_t NH = (size_t)M * HIDDEN;

  float* h     = (float*)d_ws;
  float* xl    = h + NH;
  float* xr    = xl + NH;
  float* accum = xr + NH;
  float* exbuf = accum + NH;                                    // E' * 8 floats
  unsigned* menc = (unsigned*)(exbuf + (size_t)Etot * HEADS);   // N * 8
  float* ssum  = (float*)(menc + (size_t)M * HEADS);            // N * 8

  const int gemmBlocks       = M / 16;                      // 1250
  const int nodeBlocks       = (M * 32 + 255) / 256;        // wave per node
  const int edgeWaveBlocks   = (Etot * 32 + 255) / 256;     // wave per edge
  const int edgeThreadBlocks = (Etot * HEADS + 255) / 256;  // thread per (edge,head)
  const int initBlocks       = (int)((NH + 255) / 256);

  // input projection (xl reused as raw-GEMM temp), then LN + ReLU -> h
  gemm_bias_f32wmma<<<gemmBlocks, 256, 0, stream>>>(x, W_in, b_in, xl, M, 512);
  ln_relu_node<<<nodeBlocks, 256, 0, stream>>>(xl, lng, lnb, h, M);

  for (int layer = 0; layer < 2; ++layer) {
    const float* Wl_i = Wl + (size_t)layer * HIDDEN * HIDDEN;
    const float* Wr_i = Wr + (size_t)layer * HIDDEN * HIDDEN;
    gemm_bias_f32wmma<<<gemmBlocks, 256, 0, stream>>>(h, Wl_i, bl + layer * HIDDEN, xl, M, HIDDEN);
    gemm_bias_f32wmma<<<gemmBlocks, 256, 0, stream>>>(h, Wr_i, br + layer * HIDDEN, xr, M, HIDDEN);
    init_seg<<<initBlocks, 256, 0, stream>>>(accum, menc, ssum);
    edge_logits<<<edgeWaveBlocks, 256, 0, stream>>>(ei, xl, xr, att + layer * HIDDEN,
                                                    exbuf, menc, E);
    edge_exp<<<edgeThreadBlocks, 256, 0, stream>>>(ei, exbuf, menc, ssum, E);
    edge_scatter<<<edgeWaveBlocks, 256, 0, stream>>>(ei, xl, exbuf, ssum, accum, E);
    node_finalize<<<nodeBlocks, 256, 0, stream>>>(accum, gbias + layer * HIDDEN,
                                                  ng + layer * HIDDEN, nb + layer * HIDDEN,
                                                  h, (layer == 1) ? (float*)d_out : h, M);
  }
}